// CosSimAttention_326417515271
// MI455X (gfx1250) — compile-verified
//
#include <hip/hip_runtime.h>
#include <hip/hip_bf16.h>
#include <math.h>

// ---------------------------------------------------------------------------
// CosSim attention, MI455X (gfx1250), fused flash-style single pass over xs.
//   B=64, S=8192, D=128 fp32.  xs = 256MB -> bandwidth bound (~11us @23.3TB/s).
//   Dot products run on V_WMMA_F32_16X16X4_F32 (free under the BW roofline,
//   keeps fp32 precision).  Online softmax per 1024-row chunk, combined in a
//   tiny second pass.
// Round-3: explicit A-fragment prefetch array (forces 32 loads in flight
//   regardless of scheduler heuristics); B fragments back to per-step LDS
//   reads (the round-2 register hoist collapsed load pipelining).
// ---------------------------------------------------------------------------

#define Bb 64
#define Ss 8192
#define Dd 128
#define Cc 8                    // chunks per batch row
#define CHUNK (Ss / Cc)         // 1024 s-rows per workgroup
#define NW 8                    // waves per block (wave32)
#define BLOCK (NW * 32)         // 256 threads
#define TPW (CHUNK / (NW * 16)) // 8 tiles of 16 rows per wave
#define EPSF 1e-8f

typedef float v2f __attribute__((ext_vector_type(2)));
typedef float v8f __attribute__((ext_vector_type(8)));

#if __has_builtin(__builtin_amdgcn_wave_barrier)
#define TILE_SYNC() __builtin_amdgcn_wave_barrier()
#else
#define TILE_SYNC() __syncthreads()
#endif

__global__ __launch_bounds__(BLOCK) void cosattn_pass1(
    const float* __restrict__ kq, const float* __restrict__ xs,
    const int* __restrict__ mask, float* __restrict__ cosws,
    float* __restrict__ stats, float* __restrict__ paws)
{
    __shared__ float kn[Dd];        // normalized query
    __shared__ float red[Dd];       // reduction scratch
    __shared__ float pa_sh[Dd];     // chunk partial attn accumulator
    __shared__ float l_sh;          // chunk partial denominator
    __shared__ float inv_sh;
    __shared__ float mw_sh[NW];     // per-wave running max
    __shared__ float dotb[NW][16];  // per-wave staging: row dots
    __shared__ float sqb[NW][16];   // per-wave staging: row |x|^2
    __shared__ float cosb[NW][16];  // per-wave staging: masked cos

    const int b    = blockIdx.y;
    const int c    = blockIdx.x;
    const int t    = threadIdx.x;
    const int lane = t & 31;
    const int w    = t >> 5;
    const int half = (lane < 16) ? 0 : 1;
    const int row  = lane & 15;
    const float NEG_INF = -__builtin_inff();

    // ---- normalize k[b] into LDS ----
    if (t < Dd) { float kv = kq[b * Dd + t]; red[t] = kv * kv; pa_sh[t] = 0.f; }
    if (t == 0) l_sh = 0.f;
    __syncthreads();
    if (t == 0) {
        float s = 0.f;
        for (int i = 0; i < Dd; ++i) s += red[i];
        inv_sh = 1.f / fmaxf(sqrtf(s), EPSF);
    }
    __syncthreads();
    if (t < Dd) kn[t] = kq[b * Dd + t] * inv_sh;
    __syncthreads();

    // WMMA fragment column mapping (ISA 7.12.2):
    //  A 16x4 f32: lanes 0-15 -> K=0,1 ; lanes 16-31 -> K=2,3 (M = lane%16)
    //  B 4x16 f32: VGPR0: K=0 (lanes 0-15) / K=1 (16-31); VGPR1: K=2 / K=3
    const int aoff = half ? 2 : 0;
    const int bo0  = half ? 1 : 0;
    const int bo1  = half ? 3 : 2;

    float m_w = NEG_INF, l_w = 0.f;
    float4 pav = make_float4(0.f, 0.f, 0.f, 0.f);   // d = 4*lane .. 4*lane+3
    const size_t baseBS = (size_t)b * Ss;

    for (int tt = 0; tt < TPW; ++tt) {
        const int stile = c * CHUNK + (tt * NW + w) * 16;
        const float* rowp = xs + (baseBS + stile + row) * (size_t)Dd + aoff;

        // Phase A: issue all 32 independent A-fragment loads up front
        // (immediate offsets off one base -> clause, 32-deep MLP per wave).
        v2f ar[32];
        #pragma unroll
        for (int ks = 0; ks < 32; ++ks) {
            ar[ks] = *(const v2f*)(rowp + 4 * ks);
        }

        // Phase B: 32-step WMMA chain; B fragments read from LDS (cheap,
        // short-latency, keeps persistent register pressure low).
        v8f acc = {0.f, 0.f, 0.f, 0.f, 0.f, 0.f, 0.f, 0.f};
        float sq = 0.f;
        #pragma unroll
        for (int ks = 0; ks < 32; ++ks) {
            v2f bf;
            bf.x = kn[4 * ks + bo0];   // replicated kn -> every column = row dot
            bf.y = kn[4 * ks + bo1];
            acc = __builtin_amdgcn_wmma_f32_16x16x4_f32(
                false, ar[ks], false, bf, (short)0, acc, false, false);
            sq += ar[ks].x * ar[ks].x + ar[ks].y * ar[ks].y;
        }
        // lane L covers cols {0,1} mod 4, lane L^16 covers {2,3} -> full |row|^2
        sq += __shfl_xor(sq, 16, 32);

        // stage: C/D layout -> VGPR r on lanes 0-15 = row r, lanes 16-31 = row r+8
        if (lane == 0) {
            #pragma unroll
            for (int r = 0; r < 8; ++r) dotb[w][r] = acc[r];
        }
        if (lane == 16) {
            #pragma unroll
            for (int r = 0; r < 8; ++r) dotb[w][8 + r] = acc[r];
        }
        if (lane < 16) sqb[w][lane] = sq;
        TILE_SYNC();   // per-wave staging; LDS is in-order within a wave

        if (lane < 16) {
            const int s = stile + lane;
            float cv = dotb[w][lane] / fmaxf(sqrtf(sqb[w][lane]), EPSF);
            if (mask[baseBS + s] == 0) cv = NEG_INF;
            cosb[w][lane] = cv;
            cosws[baseBS + s] = cv;     // needed for a_cos output in pass 2
        }
        TILE_SYNC();

        // ---- online softmax update (tile just read: xs re-hits L0/L2) ----
        float tmax = NEG_INF;
        #pragma unroll
        for (int i = 0; i < 16; ++i) tmax = fmaxf(tmax, cosb[w][i]);
        float m_new = fmaxf(m_w, tmax);
        if (m_new != NEG_INF) {
            float scale = (m_w == NEG_INF) ? 0.f : __expf(m_w - m_new);
            l_w *= scale;
            pav.x *= scale; pav.y *= scale; pav.z *= scale; pav.w *= scale;
            #pragma unroll
            for (int i = 0; i < 16; ++i) {
                float wv = __expf(cosb[w][i] - m_new);   // masked rows -> 0
                l_w += wv;
                const float4* xr4 = (const float4*)(xs + (baseBS + stile + i) * (size_t)Dd);
                float4 xv = xr4[lane];                   // 512B/row, one b128/lane
                pav.x += wv * xv.x;
                pav.y += wv * xv.y;
                pav.z += wv * xv.z;
                pav.w += wv * xv.w;
            }
            m_w = m_new;
        }
        TILE_SYNC();   // protect staging buffers before next tile
    }

    // ---- combine the 8 waves of this chunk ----
    if (lane == 0) mw_sh[w] = m_w;
    __syncthreads();
    float m_c = NEG_INF;
    #pragma unroll
    for (int i = 0; i < NW; ++i) m_c = fmaxf(m_c, mw_sh[i]);
    float scale = (m_w == NEG_INF) ? 0.f : __expf(m_w - m_c);
    if (lane == 0) atomicAdd(&l_sh, l_w * scale);
    atomicAdd(&pa_sh[4 * lane + 0], pav.x * scale);
    atomicAdd(&pa_sh[4 * lane + 1], pav.y * scale);
    atomicAdd(&pa_sh[4 * lane + 2], pav.z * scale);
    atomicAdd(&pa_sh[4 * lane + 3], pav.w * scale);
    __syncthreads();

    const int sc = b * Cc + c;
    if (t == 0) { stats[sc * 2] = m_c; stats[sc * 2 + 1] = l_sh; }
    if (t < Dd) paws[sc * Dd + t] = pa_sh[t];
}

__global__ __launch_bounds__(256) void cosattn_pass2(
    const float* __restrict__ cosws, const float* __restrict__ stats,
    const float* __restrict__ paws, float* __restrict__ out)
{
    __shared__ float M_sh, Z_sh;
    const int b = blockIdx.x;
    const int t = threadIdx.x;
    const float NEG_INF = -__builtin_inff();

    if (t == 0) {
        float M = NEG_INF;
        for (int ci = 0; ci < Cc; ++ci) M = fmaxf(M, stats[(b * Cc + ci) * 2]);
        float Z = 0.f;
        for (int ci = 0; ci < Cc; ++ci) {
            float m = stats[(b * Cc + ci) * 2];
            float l = stats[(b * Cc + ci) * 2 + 1];
            Z += (m == NEG_INF) ? 0.f : l * __expf(m - M);
        }
        M_sh = M; Z_sh = Z;
    }
    __syncthreads();
    const float M = M_sh, Z = Z_sh;
    const float invZ = (Z > 0.f) ? (1.f / Z) : 0.f;

    if (t < Dd) {
        float a = 0.f;
        for (int ci = 0; ci < Cc; ++ci) {
            float m = stats[(b * Cc + ci) * 2];
            float s = (m == NEG_INF) ? 0.f : __expf(m - M);
            a += paws[(b * Cc + ci) * Dd + t] * s;
        }
        out[b * Dd + t] = a * invZ;
    }

    float* acos_out = out + Bb * Dd;
    for (int it = 0; it < Ss / 256; ++it) {
        const int s = it * 256 + t;
        const float cv = cosws[(size_t)b * Ss + s];
        float a = 0.f;
        if (Z > 0.f) a = __expf(cv - M) * invZ;   // cv=-inf -> exp=0
        acos_out[(size_t)b * Ss + s] = a;
    }
}

extern "C" void kernel_launch(void* const* d_in, const int* in_sizes, int n_in,
                              void* d_out, int out_size, void* d_ws, size_t ws_size,
                              hipStream_t stream) {
    const float* kq   = (const float*)d_in[0];   // [B,D]
    const float* xs   = (const float*)d_in[1];   // [B,S,D]
    const int*   mask = (const int*)d_in[2];     // [B,S]
    float* out = (float*)d_out;                  // attn[B,D] ++ a_cos[B,S]

    float* wsf   = (float*)d_ws;
    float* cosws = wsf;                                  // B*S floats (2 MB)
    float* stats = wsf + (size_t)Bb * Ss;                // B*C*2 floats
    float* paws  = stats + Bb * Cc * 2;                  // B*C*D floats

    dim3 g1(Cc, Bb), b1(BLOCK);
    cosattn_pass1<<<g1, b1, 0, stream>>>(kq, xs, mask, cosws, stats, paws);
    cosattn_pass2<<<Bb, 256, 0, stream>>>(cosws, stats, paws, out);
}